// SpERT_60593398611972
// MI455X (gfx1250) — compile-verified
//
#include <hip/hip_runtime.h>

typedef __attribute__((ext_vector_type(2))) float v2f;
typedef __attribute__((ext_vector_type(8))) float v8f;

#define B_    4
#define L_    512
#define H_    768
#define S_    100
#define R_    100
#define E_    25
#define NE_   10
#define NR_   6
#define KP_E  800     // padded K for entity GEMM (793 -> 800, mult of 16)
#define KP_R  2368    // padded K for relation GEMM (2354 -> 2368)
#define KL_E  796     // K-loop bound (mult of 4 covering 793)
#define KL_R  2356    // K-loop bound (mult of 4 covering 2354)
#define NEGF  (-1e20f)

// ---------------------------------------------------------------------------
// Masked max-pool over contiguous spans. One block per (b, n) span.
// Step 1: reduce mask -> (start, len). Step 2: max over [start, start+len).
// Writes pooled H features to dst row; optionally appends size-emb + zero pad.
// ---------------------------------------------------------------------------
__global__ void pool_spans_kernel(const float* __restrict__ hidden,
                                  const int*   __restrict__ masks,   // (B,100,L)
                                  float*       __restrict__ dst,     // rows B*100, stride ldd
                                  int ldd,
                                  const float* __restrict__ size_emb,
                                  int do_sizes) {
  __shared__ int s_cnt[256];
  __shared__ int s_min[256];
  const int row = blockIdx.x;          // b*100 + n
  const int b   = row / 100;
  const int tid = threadIdx.x;

  const int* m = masks + (size_t)row * L_;
  int cnt = 0, mn = L_;
  for (int l = tid; l < L_; l += 256) {
    if (m[l] != 0) { cnt++; if (l < mn) mn = l; }
  }
  s_cnt[tid] = cnt; s_min[tid] = mn;
  __syncthreads();
  for (int off = 128; off > 0; off >>= 1) {
    if (tid < off) {
      s_cnt[tid] += s_cnt[tid + off];
      if (s_min[tid + off] < s_min[tid]) s_min[tid] = s_min[tid + off];
    }
    __syncthreads();
  }
  const int len = s_cnt[0];
  const int st  = s_min[0];

  const float* hb   = hidden + (size_t)b * L_ * H_;
  float*       drow = dst + (size_t)row * ldd;
  for (int h = tid; h < H_; h += 256) {
    float v = NEGF;                                  // len==0 -> -1e20 (ref semantics)
    for (int l = st; l < st + len; ++l) v = fmaxf(v, hb[(size_t)l * H_ + h]);
    drow[h] = v;
  }
  if (do_sizes && tid < 32) {                        // cols 768..799: size emb + zero pad
    int c = H_ + tid;
    if (c < KP_E) drow[c] = (tid < E_) ? size_emb[len * E_ + tid] : 0.0f;
  }
}

// ---------------------------------------------------------------------------
// Gather entity pair reprs into relation repr rows (cols 768.. of rel_reprs).
// One block per (b, r).
// ---------------------------------------------------------------------------
__global__ void gather_pairs_kernel(const float* __restrict__ ereprs,   // (400, KP_E)
                                    const int*   __restrict__ relations,// (B,R,2)
                                    float*       __restrict__ rreprs) { // (400, KP_R)
  const int row = blockIdx.x;          // b*R + r
  const int b   = row / R_;
  const int tid = threadIdx.x;
  const int e1 = relations[row * 2 + 0];
  const int e2 = relations[row * 2 + 1];
  const float* r1 = ereprs + (size_t)(b * S_ + e1) * KP_E;
  const float* r2 = ereprs + (size_t)(b * S_ + e2) * KP_E;
  float* d = rreprs + (size_t)row * KP_R;
  for (int h = tid; h < H_; h += 256) {
    d[H_ + h]     = r1[h];              // e1 hidden features
    d[2 * H_ + h] = r2[h];              // e2 hidden features
  }
  if (tid < E_) {                       // size embeddings
    d[3 * H_ + tid]      = r1[H_ + tid];
    d[3 * H_ + E_ + tid] = r2[H_ + tid];
  } else if (tid < E_ + (KP_R - (3 * H_ + 2 * E_))) {   // zero pad 2354..2367
    d[3 * H_ + E_ + tid] = 0.0f;
  }
}

// ---------------------------------------------------------------------------
// Pre-pad weights (K,N) row-major -> TRANSPOSED dense (16, Kp) so each lane's
// B-fragment pair (k, k+1) is contiguous -> single global_load_b64.
// ---------------------------------------------------------------------------
__global__ void pad_weights_kernel(const float* __restrict__ w,
                                   float* __restrict__ wpt,
                                   int K, int N, int Kp) {
  int i = blockIdx.x * blockDim.x + threadIdx.x;   // over 16 * Kp
  if (i >= 16 * Kp) return;
  int n = i / Kp, k = i - n * Kp;                  // wpt[n*Kp + k]
  wpt[i] = (k < K && n < N) ? w[k * N + n] : 0.0f;
}

// ---------------------------------------------------------------------------
// WMMA GEMM: out(400, nout) = A(400, Kp) * W(Kp, 16-padded) + bias.
// One wave (32 threads) per 16-row tile; V_WMMA_F32_16X16X4_F32 (exact f32).
// A frag: lanes 0-15: M=lane, K={k0,k0+1}; lanes 16-31: same M, K={k0+2,k0+3}.
// B frag mirrors with lane = N (weights pre-transposed to (16, Kp)).
// D: VGPR j -> row m0+j (lanes 0-15) / m0+8+j (lanes 16-31), col = lane&15.
// Both fragment loads are aligned v2f (global_load_b64).
// ---------------------------------------------------------------------------
__global__ void wmma_gemm_kernel(const float* __restrict__ A, int lda,
                                 const float* __restrict__ Wpt, int ldw,
                                 const float* __restrict__ bias,
                                 float* __restrict__ out, int nout,
                                 int kloop) {
  const int lane = threadIdx.x;               // 0..31, full wave active
  const int m0   = blockIdx.x * 16;
  const int ml   = lane & 15;
  const int kb   = (lane >> 4) << 1;          // 0 for lanes 0-15, 2 for 16-31
  const float* arow = A   + (size_t)(m0 + ml) * lda + kb;
  const float* bcol = Wpt + (size_t)ml * ldw + kb;

  v8f acc = {};
  for (int k0 = 0; k0 < kloop; k0 += 4) {
    v2f a  = *(const v2f*)(arow + k0);
    v2f bf = *(const v2f*)(bcol + k0);
    acc = __builtin_amdgcn_wmma_f32_16x16x4_f32(
        /*neg_a=*/false, a, /*neg_b=*/false, bf,
        /*c_mod=*/(short)0, acc, /*reuse_a=*/false, /*reuse_b=*/false);
  }

  const int n = ml;
  if (n < nout) {
    const float bv = bias[n];
    const int rbase = m0 + ((lane >> 4) << 3);   // +8 for upper half-wave
#pragma unroll
    for (int j = 0; j < 8; ++j) {
      out[(size_t)(rbase + j) * nout + n] = acc[j] + bv;
    }
  }
}

// ---------------------------------------------------------------------------
extern "C" void kernel_launch(void* const* d_in, const int* in_sizes, int n_in,
                              void* d_out, int out_size, void* d_ws, size_t ws_size,
                              hipStream_t stream) {
  (void)in_sizes; (void)n_in; (void)out_size; (void)ws_size;

  const float* hidden    = (const float*)d_in[0];   // (B,L,H) f32
  const int*   emasks    = (const int*)  d_in[1];   // (B,S,L) i32
  const int*   relations = (const int*)  d_in[2];   // (B,R,2) i32
  const int*   cmasks    = (const int*)  d_in[3];   // (B,R,L) i32
  const float* size_emb  = (const float*)d_in[4];   // (100,E)
  const float* w_span    = (const float*)d_in[5];   // (793,10)
  const float* b_span    = (const float*)d_in[6];   // (10,)
  const float* w_rel     = (const float*)d_in[7];   // (2354,6)
  const float* b_rel     = (const float*)d_in[8];   // (6,)
  float* out = (float*)d_out;                       // 4000 + 2400 floats

  // Workspace layout (floats)
  float* ws     = (float*)d_ws;
  float* ereprs = ws;                               // 400 * 800
  float* rreprs = ereprs + 400 * KP_E;              // 400 * 2368
  float* wpe    = rreprs + 400 * KP_R;              // 16 * 800   (transposed)
  float* wpr    = wpe    + 16 * KP_E;               // 16 * 2368  (transposed)

  // 1) Entity span max-pool + size emb + pad -> entity reprs (400, 800)
  pool_spans_kernel<<<B_ * S_, 256, 0, stream>>>(hidden, emasks, ereprs, KP_E,
                                                 size_emb, 1);
  // 2) Relation context max-pool -> rel reprs cols [0,768)
  pool_spans_kernel<<<B_ * R_, 256, 0, stream>>>(hidden, cmasks, rreprs, KP_R,
                                                 size_emb, 0);
  // 3) Gather entity pairs + sizes into rel reprs cols [768, 2354) + pad
  gather_pairs_kernel<<<B_ * R_, 256, 0, stream>>>(ereprs, relations, rreprs);
  // 4) Pad + transpose weights to (16, Kp)
  pad_weights_kernel<<<(16 * KP_E + 255) / 256, 256, 0, stream>>>(w_span, wpe, 793, NE_, KP_E);
  pad_weights_kernel<<<(16 * KP_R + 255) / 256, 256, 0, stream>>>(w_rel, wpr, 2354, NR_, KP_R);
  // 5) WMMA classifier GEMMs (25 tiles of 16 rows each)
  wmma_gemm_kernel<<<25, 32, 0, stream>>>(ereprs, KP_E, wpe, KP_E, b_span, out, NE_, KL_E);
  wmma_gemm_kernel<<<25, 32, 0, stream>>>(rreprs, KP_R, wpr, KP_R, b_rel, out + B_ * S_ * NE_, NR_, KL_R);
}